// DynamicMaskHead_90718299226442
// MI455X (gfx1250) — compile-verified
//
#include <hip/hip_runtime.h>

// CDNA5 / gfx1250: wave32, WMMA f32 16x16x4.
typedef __attribute__((ext_vector_type(2))) float v2f;
typedef __attribute__((ext_vector_type(8))) float v8f;

#define NUM_INST 128
#define CIN 10
#define HW 25600                         // 160*160 pixels per instance
#define TILES_PER_INST 1600              // HW / 16
#define TPW 16                           // tiles (of 16 px) per wave
#define WAVES_PER_INST (TILES_PER_INST / TPW)   // 100
#define WPB 8                            // waves per 256-thread block
#define NBLOCKS (NUM_INST * WAVES_PER_INST / WPB)  // 1600

__device__ __forceinline__ float relu1(float x) {
    // single v_med3_f32: median(x, 0, +inf) == max(x, 0)
    return __builtin_amdgcn_fmed3f(x, 0.0f, __builtin_inff());
}

__global__ __launch_bounds__(256, 2)
void DynamicMaskHead_kernel(const float* __restrict__ feat,
                            const float* __restrict__ prm_all,
                            float* __restrict__ out)
{
    const int lane  = threadIdx.x & 31;
    const int wave  = blockIdx.x * WPB + (threadIdx.x >> 5);
    const int n     = wave / WAVES_PER_INST;       // instance id
    const int chunk = wave % WAVES_PER_INST;       // pixel-chunk within instance
    if (n >= NUM_INST) return;

    const float* __restrict__ prm = prm_all + n * 169;
    const float* __restrict__ x   = feat + (size_t)n * CIN * HW;
    float*       __restrict__ o   = out  + (size_t)n * HW;

    const int  row  = lane & 15;     // A-matrix row / pixel-within-tile (N)
    const bool hi   = lane >= 16;    // upper half-wave holds K+2 slots
    const int  koff = hi ? 2 : 0;

    // ---------------- Per-wave weight setup (A-matrices, held in VGPRs) -----
    // Layer0 A: W0ext 16x12  (rows 8..15 = 0, col 10 = b0, col 11 = 0)
    // Layer1 A: W1ext 16x12  (rows 8..15 = 0, col 8  = b1, cols 9..11 = 0)
    // A layout (16x4 f32): lane holds row (lane&15); reg v of chunk c holds
    //   K = c*4 + v + (lane>=16 ? 2 : 0)
    v2f A0[3], A1[3];
#pragma unroll
    for (int c = 0; c < 3; ++c) {
#pragma unroll
        for (int v = 0; v < 2; ++v) {
            const int k = c * 4 + v + koff;
            float w0 = 0.0f, w1 = 0.0f;
            if (row < 8) {
                if (k < 10)       w0 = prm[row * 10 + k];      // w0 [8x10]
                else if (k == 10) w0 = prm[152 + row];         // b0
                if (k < 8)        w1 = prm[80 + row * 8 + k];  // w1 [8x8]
                else if (k == 8)  w1 = prm[160 + row];         // b1
            }
            A0[c][v] = w0;
            A1[c][v] = w1;
        }
    }

    float w2[8];
#pragma unroll
    for (int r = 0; r < 8; ++r) w2[r] = prm[144 + r];          // w2 [1x8]
    const float b2 = prm[168];

    // Constant B-chunk2 for layer 1: K=8 is the ones (bias) row, K=9..11 zero.
    v2f Cones;
    Cones[0] = hi ? 0.0f : 1.0f;
    Cones[1] = 0.0f;

    const int permIdx = row << 2;                 // ds_bpermute byte address
    const int pix0    = chunk * (TPW * 16) + row; // first pixel for this lane

    // Two independent 16-pixel tiles per iteration -> two independent WMMA
    // chains for the scheduler to interleave (hides XDL + LDS-permute latency).
    for (int t = 0; t < TPW; t += 2) {
        const int pa = pix0 + t * 16;
        const int pb = pa + 16;

        // ---- Layer 0 B-tiles (all loads unconditional: hi lanes re-read the
        // same cachelines as lo lanes; constants merged with v_cndmask).
        v2f B0a, B1a, B2a, B0b, B1b, B2b;
        B0a[0] = x[(size_t)(0 + koff) * HW + pa];      // K=0 / K=2
        B0a[1] = x[(size_t)(1 + koff) * HW + pa];      // K=1 / K=3
        B1a[0] = x[(size_t)(4 + koff) * HW + pa];      // K=4 / K=6
        B1a[1] = x[(size_t)(5 + koff) * HW + pa];      // K=5 / K=7
        float l8a = x[(size_t)8 * HW + pa];
        float l9a = x[(size_t)9 * HW + pa];
        B0b[0] = x[(size_t)(0 + koff) * HW + pb];
        B0b[1] = x[(size_t)(1 + koff) * HW + pb];
        B1b[0] = x[(size_t)(4 + koff) * HW + pb];
        B1b[1] = x[(size_t)(5 + koff) * HW + pb];
        float l8b = x[(size_t)8 * HW + pb];
        float l9b = x[(size_t)9 * HW + pb];
        B2a[0] = hi ? 1.0f : l8a;                      // K=8 / K=10 (ones)
        B2a[1] = hi ? 0.0f : l9a;                      // K=9 / K=11 (zero)
        B2b[0] = hi ? 1.0f : l8b;
        B2b[1] = hi ? 0.0f : l9b;

        // ---- Layer 0 GEMMs (two independent accumulator chains)
        v8f ya = {}, yb = {};
        ya = __builtin_amdgcn_wmma_f32_16x16x4_f32(false, A0[0], false, B0a,
                                                   (short)0, ya, false, false);
        yb = __builtin_amdgcn_wmma_f32_16x16x4_f32(false, A0[0], false, B0b,
                                                   (short)0, yb, false, false);
        ya = __builtin_amdgcn_wmma_f32_16x16x4_f32(false, A0[1], false, B1a,
                                                   (short)0, ya, false, false);
        yb = __builtin_amdgcn_wmma_f32_16x16x4_f32(false, A0[1], false, B1b,
                                                   (short)0, yb, false, false);
        ya = __builtin_amdgcn_wmma_f32_16x16x4_f32(false, A0[2], false, B2a,
                                                   (short)0, ya, false, false);
        yb = __builtin_amdgcn_wmma_f32_16x16x4_f32(false, A0[2], false, B2b,
                                                   (short)0, yb, false, false);

#pragma unroll
        for (int r = 0; r < 8; ++r) ya[r] = relu1(ya[r]);
#pragma unroll
        for (int r = 0; r < 8; ++r) yb[r] = relu1(yb[r]);

        // ---- Re-stripe D(16x16) -> B(4x16 chunks) for layer 1.
        // Lower half-wave already matches (row r in VGPR r, N=lane); upper
        // half pulls row r+2 from lanes 0-15 via ds_bpermute.
        v2f C0a, C1a, C0b, C1b;
        {
            float h;
            h = __int_as_float(__builtin_amdgcn_ds_bpermute(permIdx, __float_as_int(ya[2])));
            C0a[0] = hi ? h : ya[0];
            h = __int_as_float(__builtin_amdgcn_ds_bpermute(permIdx, __float_as_int(ya[3])));
            C0a[1] = hi ? h : ya[1];
            h = __int_as_float(__builtin_amdgcn_ds_bpermute(permIdx, __float_as_int(ya[6])));
            C1a[0] = hi ? h : ya[4];
            h = __int_as_float(__builtin_amdgcn_ds_bpermute(permIdx, __float_as_int(ya[7])));
            C1a[1] = hi ? h : ya[5];
            h = __int_as_float(__builtin_amdgcn_ds_bpermute(permIdx, __float_as_int(yb[2])));
            C0b[0] = hi ? h : yb[0];
            h = __int_as_float(__builtin_amdgcn_ds_bpermute(permIdx, __float_as_int(yb[3])));
            C0b[1] = hi ? h : yb[1];
            h = __int_as_float(__builtin_amdgcn_ds_bpermute(permIdx, __float_as_int(yb[6])));
            C1b[0] = hi ? h : yb[4];
            h = __int_as_float(__builtin_amdgcn_ds_bpermute(permIdx, __float_as_int(yb[7])));
            C1b[1] = hi ? h : yb[5];
        }

        // ---- Layer 1 GEMMs (two independent chains)
        v8f za = {}, zb = {};
        za = __builtin_amdgcn_wmma_f32_16x16x4_f32(false, A1[0], false, C0a,
                                                   (short)0, za, false, false);
        zb = __builtin_amdgcn_wmma_f32_16x16x4_f32(false, A1[0], false, C0b,
                                                   (short)0, zb, false, false);
        za = __builtin_amdgcn_wmma_f32_16x16x4_f32(false, A1[1], false, C1a,
                                                   (short)0, za, false, false);
        zb = __builtin_amdgcn_wmma_f32_16x16x4_f32(false, A1[1], false, C1b,
                                                   (short)0, zb, false, false);
        za = __builtin_amdgcn_wmma_f32_16x16x4_f32(false, A1[2], false, Cones,
                                                   (short)0, za, false, false);
        zb = __builtin_amdgcn_wmma_f32_16x16x4_f32(false, A1[2], false, Cones,
                                                   (short)0, zb, false, false);

        // ---- Layer 2 (M=1): ReLU then per-lane dot with wave-uniform w2.
        // Lanes 0-15 hold Z[r][lane] for r=0..7 across the 8 D regs.
        float accA = b2, accB = b2;
#pragma unroll
        for (int r = 0; r < 8; ++r) {
            accA = fmaf(w2[r], relu1(za[r]), accA);
            accB = fmaf(w2[r], relu1(zb[r]), accB);
        }

        if (!hi) {
            o[pa] = accA;    // 16 lanes x 4B = coalesced 64B store
            o[pb] = accB;
        }
    }
}

extern "C" void kernel_launch(void* const* d_in, const int* in_sizes, int n_in,
                              void* d_out, int out_size, void* d_ws, size_t ws_size,
                              hipStream_t stream) {
    (void)in_sizes; (void)n_in; (void)d_ws; (void)ws_size; (void)out_size;
    const float* feat   = (const float*)d_in[0];   // [1, 128*10, 160, 160] f32
    const float* params = (const float*)d_in[1];   // [128, 169] f32
    float* out = (float*)d_out;                    // [128, 1, 160, 160] f32

    dim3 grid(NBLOCKS), block(256);
    DynamicMaskHead_kernel<<<grid, block, 0, stream>>>(feat, params, out);
}